// MedianBlur_20272245637744
// MI455X (gfx1250) — compile-verified
//
#include <hip/hip_runtime.h>
#include <cstdint>
#include <cstddef>

// ---------------------------------------------------------------------------
// 3x3 median blur, zero padding, x:[8,32,512,512] f32 -> same shape.
// Memory-bound: 268 MB in + 268 MB out @ 23.3 TB/s ~= 23 us floor.
//   * one block = 128x32 output tile (1024 threads = 32 wave32s)
//     halo overhead (130*34)/(128*32) = 1.079 (vs 1.27 for 128x8)
//   * stage halo tile into LDS with gfx1250 async global->LDS DMA:
//     16B-aligned-in-LDS b128 chunks for the interior, per-column b32 / DS
//     zero-fill at image edges; s_wait_asynccnt + barrier before compute
//   * each thread computes 4 horizontal pixels via the column-sort
//     median-of-9 network (per-column sorted triples shared across outputs)
//   * nontemporal b128 stores (output is never re-read; protect L2 for halo)
// ---------------------------------------------------------------------------

#define TW 128            // tile width  (output pixels)
#define TH 32             // tile height (output pixels)
#define SW (TW + 2)       // staged width  (130)
#define SH (TH + 2)       // staged height (34)
#define LSTRIDE 132       // LDS row stride in floats (16B-aligned rows)
#define NCHUNK 33         // 4-col chunks per staged row (33*4 = 132 >= 130)
#define NTHREADS 1024     // 32 wave32s

typedef float v4f __attribute__((ext_vector_type(4)));   // native vector for NT store

__device__ __forceinline__ float f_min(float a, float b) { return fminf(a, b); }
__device__ __forceinline__ float f_max(float a, float b) { return fmaxf(a, b); }

// sorted triple: lo <= mid <= hi  (6 min/max ops)
__device__ __forceinline__ void sort3(float a, float b, float c,
                                      float& lo, float& mid, float& hi) {
    float mn = f_min(a, b);
    float mx = f_max(a, b);
    lo  = f_min(mn, c);
    float t = f_max(mn, c);
    mid = f_min(mx, t);
    hi  = f_max(mx, t);
}
__device__ __forceinline__ float med3(float a, float b, float c) {
    float mn = f_min(a, b);
    float mx = f_max(a, b);
    return f_min(mx, f_max(mn, c));
}
__device__ __forceinline__ float max3(float a, float b, float c) {
    return f_max(f_max(a, b), c);
}
__device__ __forceinline__ float min3(float a, float b, float c) {
    return f_min(f_min(a, b), c);
}

__device__ __forceinline__ void async_copy_b128(const float* gp, float* lp) {
    const uint32_t lds_off = (uint32_t)(uintptr_t)lp;   // LDS byte offset
    asm volatile("global_load_async_to_lds_b128 %0, %1, off"
                 :: "v"(lds_off), "v"(gp) : "memory");
}
__device__ __forceinline__ void async_copy_b32(const float* gp, float* lp) {
    const uint32_t lds_off = (uint32_t)(uintptr_t)lp;
    asm volatile("global_load_async_to_lds_b32 %0, %1, off"
                 :: "v"(lds_off), "v"(gp) : "memory");
}

__global__ __launch_bounds__(NTHREADS)
void median3x3_kernel(const float* __restrict__ x, float* __restrict__ out,
                      int H, int W) {
    __shared__ __align__(16) float tile[SH * LSTRIDE];

    const int n   = blockIdx.z;                 // image plane (b*c)
    const int gx0 = blockIdx.x * TW - 1;        // global col of LDS col 0
    const int gy0 = blockIdx.y * TH - 1;        // global row of LDS row 0
    const int t   = threadIdx.x;

    const float* __restrict__ img = x + (size_t)n * (size_t)H * (size_t)W;

    // ---- stage halo tile into LDS via async DMA --------------------------
    // items: (row lr, chunk k) ; chunk k covers LDS cols 4k..4k+3
    for (int i = t; i < SH * NCHUNK; i += NTHREADS) {
        const int lr = i / NCHUNK;
        const int k  = i - lr * NCHUNK;
        const int gr = gy0 + lr;
        const bool rowOK = (unsigned)gr < (unsigned)H;
        const int lc0 = k << 2;
        const int gc0 = gx0 + lc0;
        float* lp0 = &tile[lr * LSTRIDE + lc0];         // 16B aligned

        if (rowOK && gc0 >= 0 && (gc0 + 3) < W && (lc0 + 3) < SW) {
            // fast path: one async b128 (LDS 16B-aligned; global dword-aligned)
            async_copy_b128(img + (size_t)gr * (size_t)W + (size_t)gc0, lp0);
        } else {
            // edge / tail chunk: per-column handling
#pragma unroll
            for (int j = 0; j < 4; ++j) {
                const int lc = lc0 + j;
                if (lc >= SW) break;                    // LDS padding cols
                const int gc = gc0 + j;
                float* lp = lp0 + j;
                if (rowOK && (unsigned)gc < (unsigned)W) {
                    async_copy_b32(img + (size_t)gr * (size_t)W + (size_t)gc, lp);
                } else {
                    *lp = 0.0f;                         // zero padding
                }
            }
        }
    }
    asm volatile("s_wait_asynccnt 0x0" ::: "memory");
    __syncthreads();

    // ---- compute: each thread -> 4 contiguous output pixels in one row ----
    const int tx   = t & 31;       // 32 groups of 4 cols -> 128 wide
    const int ty   = t >> 5;       // 0..31 output rows
    const int lcol = tx << 2;

    const float* r0 = &tile[(ty + 0) * LSTRIDE + lcol];
    const float* r1 = &tile[(ty + 1) * LSTRIDE + lcol];
    const float* r2 = &tile[(ty + 2) * LSTRIDE + lcol];

    float a[6], b[6], c[6];
    {
        const float4 v0 = *(const float4*)r0;
        const float2 w0 = *(const float2*)(r0 + 4);
        a[0]=v0.x; a[1]=v0.y; a[2]=v0.z; a[3]=v0.w; a[4]=w0.x; a[5]=w0.y;
        const float4 v1 = *(const float4*)r1;
        const float2 w1 = *(const float2*)(r1 + 4);
        b[0]=v1.x; b[1]=v1.y; b[2]=v1.z; b[3]=v1.w; b[4]=w1.x; b[5]=w1.y;
        const float4 v2 = *(const float4*)r2;
        const float2 w2 = *(const float2*)(r2 + 4);
        c[0]=v2.x; c[1]=v2.y; c[2]=v2.z; c[3]=v2.w; c[4]=w2.x; c[5]=w2.y;
    }

    float lo[6], mi[6], hi[6];
#pragma unroll
    for (int j = 0; j < 6; ++j)
        sort3(a[j], b[j], c[j], lo[j], mi[j], hi[j]);

    float r[4];
#pragma unroll
    for (int k = 0; k < 4; ++k) {
        const float L = max3(lo[k], lo[k + 1], lo[k + 2]);
        const float M = med3(mi[k], mi[k + 1], mi[k + 2]);
        const float U = min3(hi[k], hi[k + 1], hi[k + 2]);
        r[k] = med3(L, M, U);      // median of 9
    }

    const int ogx = blockIdx.x * TW + lcol;
    const int ogy = blockIdx.y * TH + ty;
    if (ogy < H && (ogx + 3) < W) {
        float* op = out + (size_t)n * (size_t)H * (size_t)W
                        + (size_t)ogy * (size_t)W + (size_t)ogx;
        // output is never re-read: nontemporal b128 store keeps L2 for halo
        v4f res = { r[0], r[1], r[2], r[3] };
        __builtin_nontemporal_store(res, (v4f*)op);
    }
}

extern "C" void kernel_launch(void* const* d_in, const int* in_sizes, int n_in,
                              void* d_out, int out_size, void* d_ws, size_t ws_size,
                              hipStream_t stream) {
    (void)in_sizes; (void)n_in; (void)out_size; (void)d_ws; (void)ws_size;

    const float* x  = (const float*)d_in[0];
    float*      out = (float*)d_out;

    // x: [8, 32, 512, 512] float32  (from the reference setup)
    const int B = 8, C = 32, H = 512, W = 512;

    dim3 block(NTHREADS, 1, 1);
    dim3 grid(W / TW, H / TH, B * C);   // 4 x 16 x 256 blocks

    hipLaunchKernelGGL(median3x3_kernel, grid, block, 0, stream, x, out, H, W);
}